// SpheresRasterizer_76828374991352
// MI455X (gfx1250) — compile-verified
//
#include <hip/hip_runtime.h>

typedef __attribute__((ext_vector_type(2))) float v2f;
typedef __attribute__((ext_vector_type(8))) float v8f;

#define IMG   256
#define KPP   8
#define MAXN  1024

// ---------------------------------------------------------------------------
// Kernel 1: transform points through world2view then proj (with w-clamp),
// precompute per-point quantities for the WMMA distance GEMM:
//   A row  = (-2*sx, -2*sy, sx^2+sy^2, 0)   (K-vector of the rank-2 expansion)
//   ZR     = (z_view, radius^2)
// Pads entries [N, MAXN) with "never inside" defaults.
// ---------------------------------------------------------------------------
__global__ void SpheresXform_kernel(const float* __restrict__ pts,
                                    const float* __restrict__ rad,
                                    const float* __restrict__ M0,   // world2view, row-major [k][l]
                                    const float* __restrict__ M1,   // proj
                                    float4* __restrict__ outA,
                                    float2* __restrict__ outZR,
                                    int N)
{
    int i = blockIdx.x * blockDim.x + threadIdx.x;
    if (i >= MAXN) return;
    if (i < N) {
        float x = pts[3*i + 0], y = pts[3*i + 1], z = pts[3*i + 2];
        float o[4];
        #pragma unroll
        for (int l = 0; l < 4; ++l)
            o[l] = x*M0[0*4+l] + y*M0[1*4+l] + z*M0[2*4+l] + M0[3*4+l];
        float w = o[3];
        w = (fabsf(w) < 1e-8f) ? 1e-8f : w;
        float vx = o[0]/w, vy = o[1]/w, vz = o[2]/w;

        float s[4];
        #pragma unroll
        for (int l = 0; l < 4; ++l)
            s[l] = vx*M1[0*4+l] + vy*M1[1*4+l] + vz*M1[2*4+l] + M1[3*4+l];
        float w2 = s[3];
        w2 = (fabsf(w2) < 1e-8f) ? 1e-8f : w2;
        float sx = s[0]/w2, sy = s[1]/w2;

        float r = rad[i];
        outA[i]  = make_float4(-2.0f*sx, -2.0f*sy, sx*sx + sy*sy, 0.0f);
        outZR[i] = make_float2(vz, r*r);
    } else {
        outA[i]  = make_float4(0.0f, 0.0f, 0.0f, 0.0f);
        outZR[i] = make_float2(-1.0f, -1.0f);   // z<=0 -> never inside
    }
}

// ---------------------------------------------------------------------------
// Kernel 2: WMMA rasterizer. Each wave owns 16 pixels (N dim). Loop over 64
// point tiles (M dim); one V_WMMA_F32_16X16X4_F32 yields a 16x16 dist^2 tile
// with |p|^2 folded into C. Lane l holds pixel n=l%16 and points m = r + 8*hi
// across the 8 accumulator regs -> lane-local top-8 insert; halves merged at
// the end with shfl_xor(16).
//
// Skip logic: a cheap float-domain conservative trigger
//   min_r max(d[r]-r2[r], -z[r]) <= 0
// feeds a wave-uniform ballot; only tiles where some lane may hit run the
// exact predicates + sorted-insert chain. The branch is post-WMMA, so EXEC
// is all-ones at every WMMA issue.
// ---------------------------------------------------------------------------
__global__ void __launch_bounds__(256)
SpheresRaster_kernel(const float4* __restrict__ gA,
                     const float2* __restrict__ gZR,
                     int*   __restrict__ outIdx,
                     float* __restrict__ outZ,
                     float* __restrict__ outD,
                     int nTiles)
{
    __shared__ float4 sA[MAXN];
    __shared__ float2 sZR[MAXN];

    const int tid = threadIdx.x;
    #pragma unroll
    for (int q = 0; q < MAXN/256; ++q) {
        sA[tid + q*256]  = gA[tid + q*256];
        sZR[tid + q*256] = gZR[tid + q*256];
    }
    __syncthreads();

    const int  lane = tid & 31;
    const int  n    = lane & 15;
    const bool hi   = lane >= 16;

    const int wave = blockIdx.x * (blockDim.x >> 5) + (tid >> 5);
    const int pix  = wave * 16 + n;          // linear pixel id, < IMG*IMG
    const int pi   = pix >> 8;               // row  (py index)
    const int pj   = pix & (IMG - 1);        // col  (px index)

    // coords[t] = (255 - 2t)/256  (linspace(1-1/256, -(1-1/256), 256))
    const float px = (255.0f - 2.0f*(float)pj) * (1.0f/256.0f);
    const float py = (255.0f - 2.0f*(float)pi) * (1.0f/256.0f);
    const float pp = px*px + py*py;

    // B matrix (4x16 f32): lanes 0-15 -> K={0,1}, lanes 16-31 -> K={2,3}
    v2f bmat;
    bmat[0] = hi ? 1.0f : px;
    bmat[1] = hi ? 0.0f : py;

    // C = |p|^2 splat so D comes out directly as dist^2
    v8f c;
    #pragma unroll
    for (int r = 0; r < 8; ++r) c[r] = pp;

    const float INF = __builtin_inff();
    float kz[KPP], kd[KPP];
    int   ki[KPP];
    #pragma unroll
    for (int s = 0; s < KPP; ++s) { kz[s] = INF; kd[s] = -1.0f; ki[s] = -1; }

    for (int t = 0; t < nTiles; ++t) {
        // A matrix (16x4 f32): lane m=l%16, lanes 0-15 -> K={0,1}, 16-31 -> K={2,3}
        float4 a4 = sA[t*16 + n];
        v2f amat;
        amat[0] = hi ? a4.z : a4.x;
        amat[1] = hi ? a4.w : a4.y;

        // D[m][n] = -2(q.p) + |q|^2 + |p|^2 = dist2(pixel n, point m)
        v8f d = __builtin_amdgcn_wmma_f32_16x16x4_f32(
            false, amat, false, bmat, (short)0, c, false, false);

        const int base = t*16 + (hi ? 8 : 0);

        float zrz[8], zrr[8];
        #pragma unroll
        for (int r = 0; r < 8; ++r) {
            float2 zr = sZR[base + r];
            zrz[r] = zr.x;
            zrr[r] = zr.y;
        }

        // conservative trigger: lane may hit iff min_r max(d-r2, -z) <= 0
        // (false positive only at z==0 / exact boundary; exact test below)
        float g = fmaxf(d[0] - zrr[0], -zrz[0]);
        #pragma unroll
        for (int r = 1; r < 8; ++r)
            g = fminf(g, fmaxf(d[r] - zrr[r], -zrz[r]));

        // wave-uniform skip: most tiles touch no pixel of this wave
        if (__any(g <= 0.0f)) {
            #pragma unroll
            for (int r = 0; r < 8; ++r) {
                bool inside = (d[r] <= zrr[r]) && (zrz[r] > 0.0f);
                float cz = inside ? zrz[r]     : INF;
                float cd = inside ? d[r]       : -1.0f;
                int   ci = inside ? (base + r) : -1;
                // stable 8-deep sorted insert (strict z compare; idx ascending)
                #pragma unroll
                for (int s = 0; s < KPP; ++s) {
                    bool take = cz < kz[s];
                    float tz = take ? kz[s] : cz;
                    float td = take ? kd[s] : cd;
                    int   ti = take ? ki[s] : ci;
                    kz[s] = take ? cz : kz[s];
                    kd[s] = take ? cd : kd[s];
                    ki[s] = take ? ci : ki[s];
                    cz = tz; cd = td; ci = ti;
                }
            }
        }
    }

    // merge the two lane-halves (each holds a sorted top-8 over disjoint points)
    float oz[KPP], od[KPP];
    int   oi[KPP];
    #pragma unroll
    for (int q = 0; q < KPP; ++q) {
        oz[q] = __shfl_xor(kz[q], 16, 32);
        od[q] = __shfl_xor(kd[q], 16, 32);
        oi[q] = __shfl_xor(ki[q], 16, 32);
    }
    #pragma unroll
    for (int q = 0; q < KPP; ++q) {
        float cz = oz[q], cd = od[q];
        int   ci = oi[q];
        #pragma unroll
        for (int s = 0; s < KPP; ++s) {
            bool take = (cz < kz[s]) || ((cz == kz[s]) && (ci < ki[s]));
            float tz = take ? kz[s] : cz;
            float td = take ? kd[s] : cd;
            int   ti = take ? ki[s] : ci;
            kz[s] = take ? cz : kz[s];
            kd[s] = take ? cd : kd[s];
            ki[s] = take ? ci : ki[s];
            cz = tz; cd = td; ci = ti;
        }
    }

    if (!hi) {
        const size_t off = (size_t)pix * KPP;
        #pragma unroll
        for (int k = 0; k < KPP; ++k) {
            bool valid = kz[k] < INF;
            outIdx[off + k] = valid ? ki[k] : -1;
            outZ[off + k]   = valid ? kz[k] : -1.0f;
            outD[off + k]   = valid ? kd[k] : -1.0f;
        }
    }
}

// ---------------------------------------------------------------------------
extern "C" void kernel_launch(void* const* d_in, const int* in_sizes, int n_in,
                              void* d_out, int out_size, void* d_ws, size_t ws_size,
                              hipStream_t stream) {
    const float* points = (const float*)d_in[0];   // (1, N, 3)
    const float* radius = (const float*)d_in[1];   // (1, N)
    const float* w2v    = (const float*)d_in[2];   // (1, 4, 4)
    const float* proj   = (const float*)d_in[3];   // (1, 4, 4)

    int N = in_sizes[1];
    if (N > MAXN) N = MAXN;
    const int nTiles = (N + 15) / 16;

    // workspace layout: float4 A[MAXN] | float2 ZR[MAXN]  (24 KB)
    float4* wsA  = (float4*)d_ws;
    float2* wsZR = (float2*)((char*)d_ws + MAXN * sizeof(float4));

    SpheresXform_kernel<<<(MAXN + 255)/256, 256, 0, stream>>>(
        points, radius, w2v, proj, wsA, wsZR, N);

    const int HWK = IMG * IMG * KPP;
    int*   outIdx = (int*)d_out;
    float* outZ   = (float*)d_out + HWK;
    float* outD   = (float*)d_out + 2 * HWK;

    // 256 threads = 8 waves = 128 pixels per block -> 512 blocks for 256x256
    SpheresRaster_kernel<<<(IMG*IMG)/128, 256, 0, stream>>>(
        wsA, wsZR, outIdx, outZ, outD, nTiles);
}